// SoftClusterGaussianHead_9998683865146
// MI455X (gfx1250) — compile-verified
//
#include <hip/hip_runtime.h>

typedef __attribute__((ext_vector_type(2))) float v2f;
typedef __attribute__((ext_vector_type(4))) float v4f;
typedef __attribute__((ext_vector_type(8))) float v8f;

#define B_ 8
#define N_ 4096
#define D_ 512
#define K_ 8
#define EPS_ 1e-6f

// ---------------------------------------------------------------------------
// Kernel 0: zero the accumulator scratch (mu, s2, Pk are contiguous) + KL slot
// ---------------------------------------------------------------------------
__global__ void zero_kernel(float* __restrict__ base, int n, float* __restrict__ kl) {
    int i = blockIdx.x * blockDim.x + threadIdx.x;
    if (i < n) base[i] = 0.0f;
    if (i == 0) *kl = 0.0f;
}

// ---------------------------------------------------------------------------
// Kernel 1: logits = x @ W + b via WMMA f32 16x16x4, softmax over K=8,
//           write probs [B,N,K] to scratch, accumulate P_k = sum_n p.
// One wave handles 16 rows of x. Grid = B*N/16 waves, 4 waves/block.
// W is staged zero-padded to 16 columns in LDS so the inner loop has NO
// predicated loads (branch-free: 1 global b64 + 2 ds b32 per WMMA).
// A tile: x[16 rows x 4 d]  (lane&15 = row; vgpr0/1 hold d-offsets {0,1} for
//         lanes 0-15 and {2,3} for lanes 16-31)
// B tile: W[4 d x 16 k] (vgpr0 rows {0,2}, vgpr1 rows {1,3}; lane&15 = k col)
// C tile: logits[16 rows x 16 k] (vgpr v: rows v / v+8; lane&15 = k col)
// ---------------------------------------------------------------------------
__global__ void __launch_bounds__(128)
logits_softmax_kernel(const float* __restrict__ x, const float* __restrict__ W,
                      const float* __restrict__ bias, float* __restrict__ probs,
                      float* __restrict__ Pk) {
    __shared__ float Wp[D_ * 16];              // zero-padded W: 32 KB

    const int tid    = threadIdx.x;
    const int lane   = tid & 31;
    const int wave   = tid >> 5;
    const int lanelo = lane & 15;
    const int hi     = lane >> 4;              // 0 or 1
    const int klane  = lanelo & 7;

    // Stage zero-padded W into LDS (each thread: 64 entries).
    for (int i = tid; i < D_ * 16; i += 128) {
        const int row = i >> 4;
        const int col = i & 15;
        Wp[i] = (col < K_) ? W[row * K_ + col] : 0.0f;
    }
    __syncthreads();

    const int tile = blockIdx.x * 4 + wave;    // 0 .. B_*N_/16 - 1
    const int b    = tile >> 8;                // tile / (N_/16)
    const int n0   = (tile & 255) << 4;        // 16-row block base

    const float* xrow = x + ((size_t)b * N_ + n0 + lanelo) * D_ + 2 * hi;
    const float* wrow = Wp + 2 * hi * 16 + lanelo;

    v8f c = {};
    #pragma unroll 4
    for (int s = 0; s < D_ / 4; ++s) {
        v2f a;
        a.x = xrow[4 * s];
        a.y = xrow[4 * s + 1];
        v2f bb;
        bb.x = wrow[4 * s * 16];
        bb.y = wrow[(4 * s + 1) * 16];
        c = __builtin_amdgcn_wmma_f32_16x16x4_f32(false, a, false, bb,
                                                  (short)0, c, false, false);
    }

    const float bk = bias[klane];              // garbage lanes never feed valid groups
    float pk_local = 0.0f;
    #pragma unroll
    for (int v = 0; v < 8; ++v) {
        // row n = n0 + v + 8*hi ; col k = lanelo (valid k<8 in lane groups 0,2)
        float lg = c[v] + bk;
        float mx = lg;
        mx = fmaxf(mx, __shfl_xor(mx, 1, 8));
        mx = fmaxf(mx, __shfl_xor(mx, 2, 8));
        mx = fmaxf(mx, __shfl_xor(mx, 4, 8));
        float e = expf(lg - mx);
        float ssum = e;
        ssum += __shfl_xor(ssum, 1, 8);
        ssum += __shfl_xor(ssum, 2, 8);
        ssum += __shfl_xor(ssum, 4, 8);
        float p = e / ssum;
        if (lanelo < K_) {
            probs[((size_t)b * N_ + n0 + v + 8 * hi) * K_ + lanelo] = p;
            pk_local += p;
        }
    }
    if (lanelo < K_) atomicAdd(&Pk[b * K_ + lanelo], pk_local);
}

// ---------------------------------------------------------------------------
// Kernel 2: mu = p^T x  and  S2 = p^T (x*x) via paired WMMAs.
// Block = 4 waves sharing one (b, 512-row n chunk), covering 4 consecutive
// 16-wide d slices. probs chunk is staged zero-padded to 16 k-cols in LDS
// (coalesced float4 loads), so the inner loop is branch-free:
// 2 ds b32 (A = p^T) + 2 coalesced global b32 (B = x) + 2 WMMAs per step.
// A tile: p^T [16 k x 4 n] (lane&15 = k row, zero for k>=8)
// B tile: x  [4 n x 16 d]  (vgpr0 rows {0,2}, vgpr1 rows {1,3}; lane&15 = d)
// ---------------------------------------------------------------------------
__global__ void __launch_bounds__(128)
moments_kernel(const float* __restrict__ x, const float* __restrict__ probs,
               float* __restrict__ mu, float* __restrict__ s2) {
    __shared__ float Pp[512 * 16];             // zero-padded probs chunk: 32 KB

    const int tid    = threadIdx.x;
    const int lane   = tid & 31;
    const int wave   = tid >> 5;
    const int lanelo = lane & 15;
    const int hi     = lane >> 4;

    // blockIdx.x in [0, 512): b (8) x chunk (8) x slice-group (8)
    const int b      = blockIdx.x >> 6;
    const int rem    = blockIdx.x & 63;
    const int chunk  = rem >> 3;               // 0..7  (n chunk of 512 rows)
    const int sgrp   = rem & 7;                // 0..7  (group of 4 d slices)
    const int nbase  = chunk * 512;
    const int slice  = sgrp * 4 + wave;        // 0..31
    const int d      = slice * 16 + lanelo;

    const float* xb = x     + (size_t)b * N_ * D_;
    const float* pb = probs + ((size_t)b * N_ + nbase) * K_;

    // Stage the probs chunk [512 x 8] into LDS padded to [512 x 16].
    const v4f zero4 = {};
    for (int r = tid; r < 512; r += 128) {
        v4f p0 = *(const v4f*)&pb[r * K_];
        v4f p1 = *(const v4f*)&pb[r * K_ + 4];
        *(v4f*)&Pp[r * 16 + 0]  = p0;
        *(v4f*)&Pp[r * 16 + 4]  = p1;
        *(v4f*)&Pp[r * 16 + 8]  = zero4;
        *(v4f*)&Pp[r * 16 + 12] = zero4;
    }
    __syncthreads();

    const float* prow = Pp + 2 * hi * 16 + lanelo;

    v8f cmu = {};
    v8f cs2 = {};
    #pragma unroll 4
    for (int t = 0; t < 512; t += 4) {
        const int n = nbase + t + 2 * hi;
        v2f a, bx, bx2;
        a.x = prow[t * 16];
        a.y = prow[t * 16 + 16];
        const float b0 = xb[(size_t)n * D_ + d];
        const float b1 = xb[(size_t)(n + 1) * D_ + d];
        bx.x  = b0;      bx.y  = b1;
        bx2.x = b0 * b0; bx2.y = b1 * b1;
        cmu = __builtin_amdgcn_wmma_f32_16x16x4_f32(false, a, false, bx,
                                                    (short)0, cmu, false, false);
        cs2 = __builtin_amdgcn_wmma_f32_16x16x4_f32(false, a, false, bx2,
                                                    (short)0, cs2, false, false);
    }
    if (hi == 0) {  // rows k = 0..7 live in vgprs 0..7 of lanes 0-15
        #pragma unroll
        for (int v = 0; v < 8; ++v) {
            const size_t idx = ((size_t)(b * K_ + v)) * D_ + d;
            atomicAdd(&mu[idx], cmu[v]);
            atomicAdd(&s2[idx], cs2[v]);
        }
    }
}

// ---------------------------------------------------------------------------
// Kernel 3: finalize. sigma = S2 + mu^2 (P - 2); z = mu + eps*sqrt(sigma+EPS);
// pooled_z = mean_k z; KL reduced to scalar at out[B_*D_].
// ---------------------------------------------------------------------------
__global__ void __launch_bounds__(256)
finalize_kernel(const float* __restrict__ mu, const float* __restrict__ s2,
                const float* __restrict__ Pk, const float* __restrict__ eps,
                float* __restrict__ out) {
    __shared__ float red[256];
    const int idx = blockIdx.x * 256 + threadIdx.x;  // 0..B_*D_-1
    const int b = idx >> 9;
    const int d = idx & (D_ - 1);

    float acc = 0.0f, klacc = 0.0f;
    #pragma unroll
    for (int k = 0; k < K_; ++k) {
        const size_t i = ((size_t)(b * K_ + k)) * D_ + d;
        const float m   = mu[i];
        const float v   = s2[i];
        const float P   = Pk[b * K_ + k];
        const float sig = v + m * m * (P - 2.0f);
        const float sp  = sig + EPS_;
        acc   += m + eps[i] * sqrtf(sp);
        klacc += 1.0f + logf(sp) - m * m - sig;
    }
    out[b * D_ + d] = acc * (1.0f / K_);

    red[threadIdx.x] = klacc * (-0.5f / (float)(B_ * K_));
    __syncthreads();
    for (int o = 128; o > 0; o >>= 1) {
        if (threadIdx.x < o) red[threadIdx.x] += red[threadIdx.x + o];
        __syncthreads();
    }
    if (threadIdx.x == 0) atomicAdd(&out[B_ * D_], red[0]);
}

// ---------------------------------------------------------------------------
extern "C" void kernel_launch(void* const* d_in, const int* in_sizes, int n_in,
                              void* d_out, int out_size, void* d_ws, size_t ws_size,
                              hipStream_t stream) {
    (void)in_sizes; (void)n_in; (void)out_size; (void)ws_size;
    const float* x    = (const float*)d_in[0];  // [B,N,D]
    const float* W    = (const float*)d_in[1];  // [D,K]
    const float* bias = (const float*)d_in[2];  // [K]
    const float* eps  = (const float*)d_in[3];  // [B,K,D]
    float* out = (float*)d_out;                 // [B*D] pooled_z ++ [1] kl

    float* probs = (float*)d_ws;                         // B*N*K   = 262144
    float* mu    = probs + (size_t)B_ * N_ * K_;         // B*K*D   = 32768
    float* s2    = mu    + (size_t)B_ * K_ * D_;         // B*K*D   = 32768
    float* Pk    = s2    + (size_t)B_ * K_ * D_;         // B*K     = 64

    const int nz = 2 * B_ * K_ * D_ + B_ * K_;           // mu+s2+Pk contiguous
    zero_kernel<<<(nz + 255) / 256, 256, 0, stream>>>(mu, nz, &out[B_ * D_]);

    // B*N/16 = 2048 waves, 4 waves/block
    logits_softmax_kernel<<<(B_ * N_ / 16) / 4, 128, 0, stream>>>(x, W, bias, probs, Pk);

    // 8 b x 8 chunks x 8 slice-groups = 512 blocks, 4 waves each
    moments_kernel<<<512, 128, 0, stream>>>(x, probs, mu, s2);

    finalize_kernel<<<(B_ * D_) / 256, 256, 0, stream>>>(mu, s2, Pk, eps, out);
}